// GATPOS_3264175145464
// MI455X (gfx1250) — compile-verified
//
#include <hip/hip_runtime.h>
#include <hip/hip_bf16.h>

#define F_IN   128
#define P_DIM  64
#define P_HID  16
#define D1     144   // F_IN + P_HID
#define H1C1   64    // 8 heads * 8 ch
#define NCLS   40
#define NEG_SLOPE 0.2f

typedef __attribute__((ext_vector_type(2))) float v2f;
typedef __attribute__((ext_vector_type(8))) float v8f;

static __host__ __device__ inline long long cdiv(long long a, long long b) { return (a + b - 1) / b; }

// ---- monotone float<->uint encoding for atomic float max ------------------
__device__ __forceinline__ unsigned enc_f(float f) {
    unsigned u = __float_as_uint(f);
    return (u & 0x80000000u) ? ~u : (u | 0x80000000u);
}
__device__ __forceinline__ float dec_f(unsigned e) {
    unsigned u = (e & 0x80000000u) ? (e & 0x7FFFFFFFu) : ~e;
    return __uint_as_float(u);
}

// ---- copy x into xp[:, 0:128] as float4 (fully coalesced b128) ------------
__global__ void __launch_bounds__(256)
k_copyx(const float* __restrict__ x, float* __restrict__ xp, int n) {
    long long tid = (long long)blockIdx.x * blockDim.x + threadIdx.x;  // one float4
    if (tid >= (long long)n * (F_IN / 4)) return;
    int node = (int)(tid >> 5);           // F_IN/4 == 32
    int q    = (int)(tid & 31);
    const float4 v = ((const float4*)(x + (size_t)node * F_IN))[q];
    ((float4*)(xp + (size_t)node * D1))[q] = v;
}

// ---- pos MLP: xp[n, 128:144] = pos[n] @ mpW + mpb -------------------------
__global__ void __launch_bounds__(256)
k_posmlp(const float* __restrict__ pos, const float* __restrict__ mpW,
         const float* __restrict__ mpb, float* __restrict__ xp, int n) {
    long long tid = (long long)blockIdx.x * blockDim.x + threadIdx.x;
    if (tid >= (long long)n * P_HID) return;
    int node = (int)(tid >> 4);
    int jj   = (int)(tid & 15);
    float s = mpb[jj];
    const float* pr = pos + (size_t)node * P_DIM;
    #pragma unroll 8
    for (int k = 0; k < P_DIM; ++k) s += pr[k] * mpW[k * P_HID + jj];
    xp[(size_t)node * D1 + F_IN + jj] = s;
}

// ---- fp32 WMMA GEMM: Hout[Np x CV] = A[Np x K] @ W[K x CV] ----------------
// M is PADDED to a multiple of 128 in the workspace (padding rows are never
// consumed downstream), so there are no row guards at all: unconditional
// loads, unconditional stores. One wave computes 16 rows x ALL columns
// (NT 16-col tiles); the A fragment (one b64 per K-step) feeds NT WMMAs.
template<int K, int CV>
__global__ void __launch_bounds__(256)
k_gemm_wmma(const float* __restrict__ A, const float* __restrict__ W,
            float* __restrict__ Hout) {
    constexpr int NT = (CV + 15) / 16;
    const int lane = threadIdx.x & 31;
    const int wave = threadIdx.x >> 5;
    const int tileM = (blockIdx.x * (blockDim.x >> 5) + wave) * 16;  // always in-range (padded)
    const int half = lane >> 4;                 // 0 => K=0,1 ; 1 => K=2,3
    const int l16  = lane & 15;
    const int row  = tileM + l16;
    const float* Arow  = A + (size_t)row * K + (half << 1);   // 8B aligned
    const float* Wbase = W + (size_t)(half << 1) * CV;

    v8f acc[NT] = {};
    #pragma unroll 2
    for (int k0 = 0; k0 < K; k0 += 4) {
        v2f a = *(const v2f*)(Arow + k0);       // one b64, reused for NT tiles
        #pragma unroll
        for (int t = 0; t < NT; ++t) {
            int col = t * 16 + l16;
            if (col >= CV) col = CV - 1;        // constant-folded except last ragged tile
            v2f b;
            b.x = Wbase[(size_t)k0 * CV + col];
            b.y = Wbase[(size_t)k0 * CV + CV + col];
            acc[t] = __builtin_amdgcn_wmma_f32_16x16x4_f32(false, a, false, b,
                                                           (short)0, acc[t], false, false);
        }
    }

    #pragma unroll
    for (int t = 0; t < NT; ++t) {
        const int col = t * 16 + l16;
        float* op = Hout + (size_t)(tileM + (half << 3)) * CV + col;
        if (((t + 1) * 16 <= CV) || (col < CV)) {   // guard only on the ragged tile
            #pragma unroll
            for (int r = 0; r < 8; ++r) op[(size_t)r * CV] = acc[t][r];
        }
    }
}

// ---- per-node attention coefficients: as/ad[n,h] = sum_c h[n,h,c]*a[h,c] --
template<int H, int C>
__global__ void __launch_bounds__(256)
k_alpha(const float* __restrict__ h, const float* __restrict__ a_src,
        const float* __restrict__ a_dst,
        float* __restrict__ as, float* __restrict__ ad, int n) {
    int tid = blockIdx.x * blockDim.x + threadIdx.x;
    if (tid >= n * H) return;
    int node = tid / H, hh = tid % H;
    const float* hr = h + (long long)node * (H * C) + hh * C;
    float s1 = 0.f, s2 = 0.f;
    #pragma unroll
    for (int c = 0; c < C; ++c) { float v = hr[c]; s1 += v * a_src[hh * C + c]; s2 += v * a_dst[hh * C + c]; }
    as[tid] = s1; ad[tid] = s2;
}

// ---- edge pass 1: segment max of leaky_relu(as[src]+ad[dst]) --------------
template<int H>
__global__ void __launch_bounds__(256)
k_edge_max(const int* __restrict__ srcs, const int* __restrict__ dsts,
           int E_, int Etot, const float* __restrict__ as,
           const float* __restrict__ ad, unsigned* __restrict__ m) {
    long long tid = (long long)blockIdx.x * blockDim.x + threadIdx.x;
    if (tid >= (long long)Etot * H) return;
    int e = (int)(tid / H), hh = (int)(tid % H);
    int s, d;
    if (e < E_) { s = srcs[e]; d = dsts[e]; } else { s = d = e - E_; }
    float ev = as[s * H + hh] + ad[d * H + hh];
    ev = ev > 0.f ? ev : ev * NEG_SLOPE;
    atomicMax(&m[d * H + hh], enc_f(ev));
}

// ---- edge pass 2: den[dst,h] += e^x ; num[dst,h,c] += e^x * h[src,h,c] ----
// One thread per (edge, head, 8-channel chunk): two b128 gathers + 8 atomics.
template<int H, int C>
__global__ void __launch_bounds__(256)
k_edge_agg(const int* __restrict__ srcs, const int* __restrict__ dsts,
           int E_, int Etot, const float* __restrict__ as,
           const float* __restrict__ ad, const unsigned* __restrict__ m,
           const float* __restrict__ hsrc,
           float* __restrict__ den, float* __restrict__ num) {
    constexpr int NCH = C / 8;                  // 8-channel chunks per head
    long long tid = (long long)blockIdx.x * blockDim.x + threadIdx.x;
    if (tid >= (long long)Etot * H * NCH) return;
    int ch = (int)(tid % NCH);
    long long t2 = tid / NCH;
    int hh = (int)(t2 % H);
    int e  = (int)(t2 / H);
    int s, d;
    if (e < E_) { s = srcs[e]; d = dsts[e]; } else { s = d = e - E_; }
    float ev = as[s * H + hh] + ad[d * H + hh];
    ev = ev > 0.f ? ev : ev * NEG_SLOPE;
    float ex = expf(ev - dec_f(m[d * H + hh]));
    if (ch == 0) atomicAdd(&den[d * H + hh], ex);
    const float* hp = hsrc + (size_t)s * (H * C) + hh * C + ch * 8;  // 16B aligned
    float*       np = num  + (size_t)d * (H * C) + hh * C + ch * 8;
    float4 v0 = *(const float4*)hp;
    float4 v1 = *(const float4*)(hp + 4);
    atomicAdd(np + 0, ex * v0.x); atomicAdd(np + 1, ex * v0.y);
    atomicAdd(np + 2, ex * v0.z); atomicAdd(np + 3, ex * v0.w);
    atomicAdd(np + 4, ex * v1.x); atomicAdd(np + 5, ex * v1.y);
    atomicAdd(np + 6, ex * v1.z); atomicAdd(np + 7, ex * v1.w);
}

// ---- finalize layer 1: elu(num/den + b1) ----------------------------------
__global__ void __launch_bounds__(256)
k_fin1(const float* __restrict__ num, const float* __restrict__ den,
       const float* __restrict__ b1, float* __restrict__ out, int n) {
    long long tid = (long long)blockIdx.x * blockDim.x + threadIdx.x;
    if (tid >= (long long)n * H1C1) return;
    int node = (int)(tid >> 6);
    int hc   = (int)(tid & 63);
    int hh   = hc >> 3;
    float v = num[tid] / den[node * 8 + hh] + b1[hc];
    out[tid] = v > 0.f ? v : (expf(v) - 1.0f);
}

// ---- finalize layer 2: num/den + b2, then log_softmax over 40 classes -----
__global__ void __launch_bounds__(256)
k_fin2(const float* __restrict__ num, const float* __restrict__ den,
       const float* __restrict__ b2, float* __restrict__ out, int n) {
    int node = blockIdx.x * blockDim.x + threadIdx.x;
    if (node >= n) return;
    float v[NCLS];
    float dinv = 1.0f / den[node];
    float mx = -3.4e38f;
    #pragma unroll
    for (int c = 0; c < NCLS; ++c) {
        v[c] = num[(long long)node * NCLS + c] * dinv + b2[c];
        mx = fmaxf(mx, v[c]);
    }
    float s = 0.f;
    #pragma unroll
    for (int c = 0; c < NCLS; ++c) s += expf(v[c] - mx);
    float lse = mx + logf(s);
    #pragma unroll
    for (int c = 0; c < NCLS; ++c) out[(long long)node * NCLS + c] = v[c] - lse;
}

extern "C" void kernel_launch(void* const* d_in, const int* in_sizes, int n_in,
                              void* d_out, int out_size, void* d_ws, size_t ws_size,
                              hipStream_t stream) {
    const float* x   = (const float*)d_in[0];
    const float* pos = (const float*)d_in[1];
    const int*   ei  = (const int*)  d_in[2];
    const float* mpW = (const float*)d_in[3];
    const float* mpb = (const float*)d_in[4];
    const float* W1  = (const float*)d_in[5];
    const float* a1s = (const float*)d_in[6];
    const float* a1d = (const float*)d_in[7];
    const float* b1  = (const float*)d_in[8];
    const float* W2  = (const float*)d_in[9];
    const float* a2s = (const float*)d_in[10];
    const float* a2d = (const float*)d_in[11];
    const float* b2  = (const float*)d_in[12];
    float* out = (float*)d_out;

    const int N  = in_sizes[0] / F_IN;
    const int Np = (int)cdiv(N, 128) * 128;   // rows padded to 128 (16 x 8 waves)
    const int E  = in_sizes[2] / 2;
    const int Etot = E + N;                   // self-loops appended
    const int* srcs = ei;
    const int* dsts = ei + E;

    // ---- workspace layout (floats), padded GEMM regions, phase reuse ----
    float* ws = (float*)d_ws;
    size_t o = 0;
    float*    xp   = ws + o;  o += (size_t)Np * D1;    // region A (reused below)
    float*    h1   = ws + o;  o += (size_t)Np * H1C1;  // region B (reused below)
    float*    as1  = ws + o;  o += (size_t)N * 8;
    float*    ad1  = ws + o;  o += (size_t)N * 8;
    unsigned* m1   = (unsigned*)(ws + o); o += (size_t)N * 8;   // m1/den1/num1 contiguous
    float*    den1 = ws + o;  o += (size_t)N * 8;
    float*    num1 = ws + o;  o += (size_t)N * H1C1;
    // phase-2 overlays (fit checked against padded region sizes):
    float*    h1a  = xp;                          // Np*64   (xp dead after GEMM1)
    float*    h2   = xp + (size_t)Np * H1C1;      // Np*40
    float*    as2  = h2 + (size_t)Np * NCLS;      // N
    float*    ad2  = as2 + (size_t)N;             // N      (64+40+~2 <= 144 per row)
    unsigned* m2   = (unsigned*)h1;               // h1 dead after fin1; m2/den2/num2 contiguous
    float*    den2 = h1 + (size_t)N;
    float*    num2 = h1 + (size_t)N * 2;          // N*40   (42 <= 64*Np/N per row)

    const int B = 256;

    // 1) pos-MLP + concat (split: coalesced copy + dedicated MLP)
    k_copyx<<<(int)cdiv((long long)N * 32, B), B, 0, stream>>>(x, xp, N);
    k_posmlp<<<(int)cdiv((long long)N * P_HID, B), B, 0, stream>>>(pos, mpW, mpb, xp, N);

    // 2) h1 = xp @ W1   (fp32 WMMA, K=144, 4 column tiles per wave, no guards)
    k_gemm_wmma<D1, H1C1><<<Np / 128, B, 0, stream>>>(xp, W1, h1);

    // 3) attention coefficients (8 heads x 8 ch)
    k_alpha<8, 8><<<(int)cdiv((long long)N * 8, B), B, 0, stream>>>(h1, a1s, a1d, as1, ad1, N);

    // 4) zero m1/den1/num1 (contiguous N*80 floats)
    hipMemsetAsync(m1, 0, (size_t)N * 80 * sizeof(float), stream);

    // 5) edge softmax (max pass + fused exp/scatter pass)
    k_edge_max<8><<<(int)cdiv((long long)Etot * 8, B), B, 0, stream>>>(srcs, dsts, E, Etot, as1, ad1, m1);
    k_edge_agg<8, 8><<<(int)cdiv((long long)Etot * 8, B), B, 0, stream>>>(
        srcs, dsts, E, Etot, as1, ad1, m1, h1, den1, num1);

    // 6) h1a = elu(num1/den1 + b1)
    k_fin1<<<(int)cdiv((long long)N * H1C1, B), B, 0, stream>>>(num1, den1, b1, h1a, N);

    // 7) h2 = h1a @ W2   (fp32 WMMA, K=64, 3 column tiles, last ragged)
    k_gemm_wmma<H1C1, NCLS><<<Np / 128, B, 0, stream>>>(h1a, W2, h2);

    // 8) attention coefficients (1 head x 40 ch)
    k_alpha<1, NCLS><<<(int)cdiv((long long)N, B), B, 0, stream>>>(h2, a2s, a2d, as2, ad2, N);

    // 9) zero m2/den2/num2 (contiguous N*42 floats)
    hipMemsetAsync(m2, 0, (size_t)N * 42 * sizeof(float), stream);

    // 10) layer-2 edge softmax (5 chunks of 8 channels)
    k_edge_max<1><<<(int)cdiv((long long)Etot, B), B, 0, stream>>>(srcs, dsts, E, Etot, as2, ad2, m2);
    k_edge_agg<1, NCLS><<<(int)cdiv((long long)Etot * 5, B), B, 0, stream>>>(
        srcs, dsts, E, Etot, as2, ad2, m2, h2, den2, num2);

    // 11) bias + log_softmax -> d_out
    k_fin2<<<(int)cdiv((long long)N, B), B, 0, stream>>>(num2, den2, b2, out, N);
}